// CTR_27754078666791
// MI455X (gfx1250) — compile-verified
//
#include <hip/hip_runtime.h>

#define SALPHA 20.0f
#define EPSV   1e-16f
#define THR    0.005f
#define BDIM   64
#define KDIM   256
#define VDIM   5000
#define VP     5120   // V padded to multiple of 256 (5000 even -> pad starts on a k-pair boundary)

typedef float v2f __attribute__((ext_vector_type(2)));
typedef float v8f __attribute__((ext_vector_type(8)));

// Pair-interleaved layouts so every WMMA fragment is ONE aligned 8-byte load:
//   uT  : element (k,b) at uT [(k>>1)*128     + b*2 + (k&1)]   (256x64)
//   vT  : element (v,b) at vT [(v>>1)*128     + b*2 + (v&1)]   (5120x64)
//   Kmat: element (k,v) at Kmat[(k>>1)*(2*VP) + v*2 + (k&1)]   (256x5120)  B-frags of GEMM1
//   WM  : same layout as Kmat (elementwise Kmat*Mt)
//   KmatT: element (v,k) at KmatT[(v>>1)*512  + k*2 + (v&1)]   (5120x256)  B-frags of GEMM2
//   bT  : element (v,b) at bT[v*64 + b]  (plain transpose; only used in epilogue, float4-friendly)

__device__ __forceinline__ float group16_sum(float s) {
  // sum across the 16-lane half-group (wave32; xor masks 1..8 stay within halves)
  s += __shfl_xor(s, 1, 32);
  s += __shfl_xor(s, 2, 32);
  s += __shfl_xor(s, 4, 32);
  s += __shfl_xor(s, 8, 32);
  return s;
}

// ---------------- init kernels ----------------

__global__ void k_init_mat(const float* __restrict__ M,
                           float* __restrict__ Kmat,
                           float* __restrict__ KmatT,
                           float* __restrict__ WM) {
  int idx = blockIdx.x * 256 + threadIdx.x;   // 0 .. 256*5120-1
  int k = idx & 255;
  int v = idx >> 8;
  float e = 0.f, wm = 0.f;
  if (v < VDIM) {
    float m = M[v * KDIM + k];
    e  = expf(-SALPHA * m);
    wm = e * m;
  }
  Kmat [(size_t)(k >> 1) * (2 * VP)   + v * 2 + (k & 1)] = e;
  KmatT[(size_t)(v >> 1) * (2 * KDIM) + k * 2 + (v & 1)] = e;
  WM   [(size_t)(k >> 1) * (2 * VP)   + v * 2 + (k & 1)] = wm;
}

// uT = 1/K everywhere; vT = 1/V (pad 0); bT[v][b] = b[b][v] (pad 0)
__global__ void k_init_T(const float* __restrict__ b_in,
                         float* __restrict__ uT,
                         float* __restrict__ vT,
                         float* __restrict__ bT) {
  int idx = blockIdx.x * 256 + threadIdx.x;
  if (idx < KDIM * BDIM) uT[idx] = 1.0f / (float)KDIM;   // constant -> layout-agnostic
  int e = idx - KDIM * BDIM;
  if (e >= 0 && e < VP * BDIM) {
    // paired layout: linear index e covers gv pairs in order; gv<5000 <=> e < 5000*64
    vT[e] = (e < VDIM * BDIM) ? (1.0f / (float)VDIM) : 0.f;
  }
  int e2 = idx - (KDIM * BDIM + VP * BDIM);
  if (e2 >= 0 && e2 < VP * BDIM) {
    int gb = e2 & 63, gv = e2 >> 6;
    bT[e2] = (gv < VDIM) ? b_in[gb * VDIM + gv] : 0.f;
  }
}

// blocks 0..63 -> a_sum[b]; blocks 64..127 -> b_sum[b]; block 0 clears done flag
__global__ void k_init_sums(const float* __restrict__ a_in,
                            const float* __restrict__ b_in,
                            float* __restrict__ a_sum,
                            float* __restrict__ b_sum,
                            int* __restrict__ ctrl) {
  __shared__ float red[256];
  int blk = blockIdx.x, tid = threadIdx.x;
  float s = 0.f;
  if (blk < 64) {
    s = a_in[blk * KDIM + tid];
  } else {
    int gb = blk - 64;
    for (int j = tid; j < VDIM; j += 256) s += b_in[gb * VDIM + j];
  }
  red[tid] = s;
  __syncthreads();
  for (int off = 128; off > 0; off >>= 1) {
    if (tid < off) red[tid] += red[tid + off];
    __syncthreads();
  }
  if (tid == 0) {
    if (blk < 64) a_sum[blk] = red[0];
    else          b_sum[blk - 64] = red[0];
    if (blk == 0) ctrl[0] = 0;
  }
}

// ---------------- GEMM1 (and loss pass) ----------------
// MODE 0: KTu = u x Kmat -> v_new = b/(KTu+eps) -> vTout; t%50==0: err_v partials.
// MODE 1: T = u x WM; loss partials sum_v vT*T per batch row.
// Block: 64 output cols, 8 waves; wave w -> n-tile (w&3), m-tiles {2*(w>>2),2*(w>>2)+1}.
// Inner loop: 3x global_load_b64 feeding 2x v_wmma_f32_16x16x4_f32.
template <int MODE>
__launch_bounds__(256)
__global__ void k_gemmA(const float* __restrict__ uT,
                        const float* __restrict__ Bmat,  // Kmat (MODE0) or WM (MODE1), paired
                        const float* __restrict__ bT,    // [VP][64] (MODE0)
                        const float* __restrict__ vTin,  // paired (MODE1)
                        float* __restrict__ vTout,       // paired (MODE0)
                        float* __restrict__ part,        // [80][64]
                        const int* __restrict__ ctrl,
                        int t) {
  if (MODE == 0 && ctrl[0]) return;
  const bool CHECK = (MODE == 1) || ((t % 50) == 0);
  __shared__ float wpart[512];
  const int tid  = threadIdx.x;
  const int w    = tid >> 5;
  const int lane = tid & 31;
  const int half = lane >> 4;
  if (CHECK) { wpart[tid] = 0.f; wpart[tid + 256] = 0.f; }
  __syncthreads();

  const int nt   = w & 3;
  const int mtb  = (w >> 2) * 2;
  const int col  = blockIdx.x * 64 + nt * 16 + (lane & 15);  // global v (0..5119)
  const int row0 = mtb * 16 + (lane & 15);                   // A-frag row, tile 0

  const v8f zero8 = {0.f, 0.f, 0.f, 0.f, 0.f, 0.f, 0.f, 0.f};
  v8f acc[2] = {zero8, zero8};

  for (int ks = 0; ks < 64; ++ks) {
    const int k2 = ks * 2 + half;      // k-pair index (covers k = 4*ks + 2*half, +1)
    v2f bf = *reinterpret_cast<const v2f*>(Bmat + (size_t)k2 * (2 * VP) + col * 2);
    v2f a0 = *reinterpret_cast<const v2f*>(uT + k2 * 128 + row0 * 2);
    v2f a1 = *reinterpret_cast<const v2f*>(uT + k2 * 128 + (row0 + 16) * 2);
    acc[0] = __builtin_amdgcn_wmma_f32_16x16x4_f32(false, a0, false, bf, (short)0, acc[0], false, false);
    acc[1] = __builtin_amdgcn_wmma_f32_16x16x4_f32(false, a1, false, bf, (short)0, acc[1], false, false);
  }

  const int gv = col;
  const size_t vpo = (size_t)(gv >> 1) * 128 + (gv & 1);  // paired v-row base for this lane
#pragma unroll
  for (int i = 0; i < 2; ++i) {
    const int gb0 = (mtb + i) * 16 + 8 * half;  // 8 consecutive batch rows per lane
    if (MODE == 0) {
      const float4* bp = reinterpret_cast<const float4*>(bT + (size_t)gv * 64 + gb0);
      float4 bv0 = bp[0], bv1 = bp[1];
      float bb[8] = {bv0.x, bv0.y, bv0.z, bv0.w, bv1.x, bv1.y, bv1.z, bv1.w};
      float vn[8];
#pragma unroll
      for (int r = 0; r < 8; ++r) {
        float c = acc[i][r];
        vn[r] = bb[r] / (c + EPSV);  // padded cols: 0/(0+eps)=0
        if (CHECK) {
          float s = group16_sum(vn[r] * c);
          if ((lane & 15) == 0) wpart[w * 64 + gb0 + r] += s;
        }
      }
#pragma unroll
      for (int r = 0; r < 8; ++r) vTout[vpo + (gb0 + r) * 2] = vn[r];
    } else {
#pragma unroll
      for (int r = 0; r < 8; ++r) {
        float vv = vTin[vpo + (gb0 + r) * 2];
        float s = group16_sum(vv * acc[i][r]);
        if ((lane & 15) == 0) wpart[w * 64 + gb0 + r] += s;
      }
    }
  }

  if (CHECK) {
    __syncthreads();
    if (tid < 64) {
      float s = 0.f;
#pragma unroll
      for (int w2 = 0; w2 < 8; ++w2) s += wpart[w2 * 64 + tid];
      part[blockIdx.x * 64 + tid] = s;
    }
  }
}

// ---------------- GEMM2 stage 1: partial[chunk] = v x KmatT over 128 v's ----------------
// Inner loop: 5x global_load_b64 feeding 4x v_wmma_f32_16x16x4_f32.
__launch_bounds__(256)
__global__ void k_gemmB1(const float* __restrict__ vT,     // paired
                         const float* __restrict__ KmatT,  // paired
                         float* __restrict__ partial,      // [40][64][256]
                         const int* __restrict__ ctrl) {
  if (ctrl[0]) return;
  const int tid = threadIdx.x, w = tid >> 5, lane = tid & 31, half = lane >> 4;
  const v8f zero8 = {0.f, 0.f, 0.f, 0.f, 0.f, 0.f, 0.f, 0.f};

  for (int nt2 = 0; nt2 < 2; ++nt2) {
    const int col = (w * 2 + nt2) * 16 + (lane & 15);  // output k index 0..255
    v8f acc[4] = {zero8, zero8, zero8, zero8};
    for (int ks = 0; ks < 32; ++ks) {
      const int k2 = blockIdx.x * 64 + ks * 2 + half;  // v-pair index
      v2f bf = *reinterpret_cast<const v2f*>(KmatT + (size_t)k2 * 512 + col * 2);
#pragma unroll
      for (int mt = 0; mt < 4; ++mt) {
        const int row = mt * 16 + (lane & 15);
        v2f a = *reinterpret_cast<const v2f*>(vT + (size_t)k2 * 128 + row * 2);
        acc[mt] = __builtin_amdgcn_wmma_f32_16x16x4_f32(false, a, false, bf, (short)0, acc[mt], false, false);
      }
    }
#pragma unroll
    for (int mt = 0; mt < 4; ++mt) {
      size_t base = (size_t)blockIdx.x * 16384 + (size_t)(mt * 16 + 8 * half) * 256 + col;
#pragma unroll
      for (int r = 0; r < 8; ++r) partial[base + (size_t)r * 256] = acc[mt][r];
    }
  }
}

// ---------------- GEMM2 stage 2: Kv = sum partials; u_new = a/(Kv+eps); err_u per b ----------------
__launch_bounds__(256)
__global__ void k_gemmB2(const float* __restrict__ partial,
                         const float* __restrict__ a_in,
                         const float* __restrict__ a_sum,
                         float* __restrict__ uT,     // paired
                         float* __restrict__ erru_b, // [64]
                         const int* __restrict__ ctrl,
                         int t) {
  if (ctrl[0]) return;
  __shared__ float red[256];
  const int gb = blockIdx.x, k = threadIdx.x;
  float s = 0.f;
  for (int c = 0; c < 40; ++c) s += partial[(size_t)c * 16384 + gb * 256 + k];
  const float un = a_in[gb * KDIM + k] / (s + EPSV);
  uT[(k >> 1) * 128 + gb * 2 + (k & 1)] = un;
  if ((t % 50) == 0) {
    red[k] = un * s;
    __syncthreads();
    for (int off = 128; off > 0; off >>= 1) {
      if (k < off) red[k] += red[k + off];
      __syncthreads();
    }
    if (k == 0) erru_b[gb] = fabsf(red[0] - a_sum[gb]);
  }
}

// ---------------- convergence check ----------------
__global__ void k_done(const float* __restrict__ errvp,  // [80][64]
                       const float* __restrict__ b_sum,
                       const float* __restrict__ erru_b,
                       int* __restrict__ ctrl) {
  if (ctrl[0]) return;
  __shared__ float m[64];
  const int tid = threadIdx.x;  // 64 threads
  float s = 0.f;
  for (int c = 0; c < 80; ++c) s += errvp[c * 64 + tid];
  m[tid] = fmaxf(fabsf(s - b_sum[tid]), erru_b[tid]);
  __syncthreads();
  for (int off = 32; off > 0; off >>= 1) {
    if (tid < off) m[tid] = fmaxf(m[tid], m[tid + off]);
    __syncthreads();
  }
  if (tid == 0 && m[0] <= THR) ctrl[0] = 1;
}

// ---------------- final loss reduce ----------------
__global__ void k_loss2(const float* __restrict__ lossp,  // [80][64]
                        float* __restrict__ out) {
  __shared__ float m[64];
  const int tid = threadIdx.x;  // 64 threads
  float s = 0.f;
  for (int c = 0; c < 80; ++c) s += lossp[c * 64 + tid];
  m[tid] = s;
  __syncthreads();
  for (int off = 32; off > 0; off >>= 1) {
    if (tid < off) m[tid] += m[tid + off];
    __syncthreads();
  }
  if (tid == 0) out[0] = m[0] * (1.0f / (float)BDIM);
}

// ---------------- host ----------------
extern "C" void kernel_launch(void* const* d_in, const int* in_sizes, int n_in,
                              void* d_out, int out_size, void* d_ws, size_t ws_size,
                              hipStream_t stream) {
  (void)in_sizes; (void)n_in; (void)out_size; (void)ws_size;
  const float* a = (const float*)d_in[0];  // [64,256]
  const float* b = (const float*)d_in[1];  // [64,5000]
  const float* M = (const float*)d_in[2];  // [5000,256]

  float* ws      = (float*)d_ws;           // needs ~21.1 MB
  float* Kmat    = ws;                                   // 256*5120 (paired)
  float* KmatT   = Kmat    + (size_t)KDIM * VP;          // 5120*256 (paired)
  float* WM      = KmatT   + (size_t)VP * KDIM;          // 256*5120 (paired)
  float* uT      = WM      + (size_t)KDIM * VP;          // 256*64   (paired)
  float* vT      = uT      + (size_t)KDIM * BDIM;        // 5120*64  (paired)
  float* bT      = vT      + (size_t)VP * BDIM;          // 5120*64
  float* partial = bT      + (size_t)VP * BDIM;          // 40*64*256
  float* errvp   = partial + (size_t)40 * 64 * 256;      // 80*64
  float* lossp   = errvp   + 80 * 64;                    // 80*64
  float* errub   = lossp   + 80 * 64;                    // 64
  float* asum    = errub   + 64;                         // 64
  float* bsum    = asum    + 64;                         // 64
  int*   ctrl    = (int*)(bsum + 64);                    // done flag

  k_init_mat <<<5120, 256, 0, stream>>>(M, Kmat, KmatT, WM);
  k_init_T   <<<2624, 256, 0, stream>>>(b, uT, vT, bT);
  k_init_sums<<<128,  256, 0, stream>>>(a, b, asum, bsum, ctrl);

  for (int t = 0; t < 200; ++t) {
    k_gemmA<0><<<80, 256, 0, stream>>>(uT, Kmat, bT, nullptr, vT, errvp, ctrl, t);
    k_gemmB1 <<<40, 256, 0, stream>>>(vT, KmatT, partial, ctrl);
    k_gemmB2 <<<64, 256, 0, stream>>>(partial, a, asum, uT, errub, ctrl, t);
    if ((t % 50) == 0) k_done<<<1, 64, 0, stream>>>(errvp, bsum, errub, ctrl);
  }

  k_gemmA<1><<<80, 256, 0, stream>>>(uT, WM, nullptr, vT, nullptr, lossp, ctrl, 0);
  k_loss2  <<<1, 64, 0, stream>>>(lossp, (float*)d_out);
}